// GraphConvolutionLayer_46067819217039
// MI455X (gfx1250) — compile-verified
//
#include <hip/hip_runtime.h>

typedef __attribute__((ext_vector_type(2))) float v2f;
typedef __attribute__((ext_vector_type(4))) float v4f;
typedef __attribute__((ext_vector_type(8))) float v8f;

#define N_NODES 50000
#define N_EDGES 800000
#define IN_F    128
#define OUT_F   128
#define EDGE_F  6

// ---------------------------------------------------------------------------
// Zero-init the accumulator (segment_sum target). Vectorized b128 stores.
// ---------------------------------------------------------------------------
__global__ void gcn_zero(float* __restrict__ out, int n4) {
    int i = blockIdx.x * blockDim.x + threadIdx.x;
    if (i < n4) {
        v4f z = {0.f, 0.f, 0.f, 0.f};
        ((v4f*)out)[i] = z;
    }
}

// ---------------------------------------------------------------------------
// T = NF @ W + b  using V_WMMA_F32_16X16X4_F32 (exact fp32 matrix core path).
// Block = 128 threads = 4 waves. Block b owns rows [16b, 16b+16).
// Wave w owns output columns [32w, 32w+32) as two 16x16 tiles (A reused).
//
// Fragment layouts (CDNA5 ISA 7.12.2):
//   A 16x4 f32 : lane l (0-15) -> M=l, lanes 16-31 -> same M; K = vgpr + 2*(l>=16)
//   B 4x16 f32 : lane l&15 -> N;  K = vgpr + 2*(l>=16)
//   C 16x16 f32: vgpr v, lane l -> M = v + 8*(l>=16), N = l&15
// ---------------------------------------------------------------------------
__global__ __launch_bounds__(128) void gcn_node_transform(
    const float* __restrict__ NF, const float* __restrict__ W,
    const float* __restrict__ b, float* __restrict__ T) {

    const int wave = threadIdx.x >> 5;   // 0..3
    const int lane = threadIdx.x & 31;
    const int g    = lane >> 4;          // lane group (adds 2 to K / 8 to M)
    const int l    = lane & 15;
    const int m0   = blockIdx.x * 16;
    const int n0   = wave * 32;

    const float bias0 = b[n0 + l];
    const float bias1 = b[n0 + 16 + l];
    v8f c0, c1;
#pragma unroll
    for (int v = 0; v < 8; ++v) { c0[v] = bias0; c1[v] = bias1; }

    const float* arow = NF + (size_t)(m0 + l) * IN_F + 2 * g;  // K offset 2g within quad

#pragma unroll 4
    for (int kt = 0; kt < IN_F / 4; ++kt) {
        const int kb = kt * 4 + 2 * g;
        v2f a = *(const v2f*)(arow + kt * 4);          // K = kb, kb+1 for this lane
        v2f b0, b1;
        b0.x = W[(size_t)(kb + 0) * OUT_F + n0 + l];
        b0.y = W[(size_t)(kb + 1) * OUT_F + n0 + l];
        b1.x = W[(size_t)(kb + 0) * OUT_F + n0 + 16 + l];
        b1.y = W[(size_t)(kb + 1) * OUT_F + n0 + 16 + l];
        c0 = __builtin_amdgcn_wmma_f32_16x16x4_f32(false, a, false, b0,
                                                   (short)0, c0, false, false);
        c1 = __builtin_amdgcn_wmma_f32_16x16x4_f32(false, a, false, b1,
                                                   (short)0, c1, false, false);
    }

#pragma unroll
    for (int v = 0; v < 8; ++v) {
        const int row = m0 + v + 8 * g;
        T[(size_t)row * OUT_F + n0 + l]      = c0[v];
        T[(size_t)row * OUT_F + n0 + 16 + l] = c1[v];
    }
}

// ---------------------------------------------------------------------------
// Per-edge: msg = T[src] + (ef @ We + be); atomically add msg into out[dst].
// One wave per edge; lane owns 4 contiguous columns (b128 gather of T[src],
// then 4 non-returning global_atomic_add_f32 into the L2-resident accumulator).
// We/be staged in LDS once per block.
// ---------------------------------------------------------------------------
__global__ __launch_bounds__(256) void gcn_edge_scatter(
    const int*   __restrict__ EI, const float* __restrict__ EF,
    const float* __restrict__ We, const float* __restrict__ be,
    const float* __restrict__ T,  float* __restrict__ out) {

    __shared__ float sWe[EDGE_F * OUT_F];
    __shared__ float sbe[OUT_F];
    for (int i = threadIdx.x; i < EDGE_F * OUT_F; i += blockDim.x) sWe[i] = We[i];
    if (threadIdx.x < OUT_F) sbe[threadIdx.x] = be[threadIdx.x];
    __syncthreads();

    const int wave = threadIdx.x >> 5;
    const int lane = threadIdx.x & 31;
    const int e    = blockIdx.x * 8 + wave;
    if (e >= N_EDGES) return;

    const int src = EI[(size_t)e * 2 + 0];
    const int dst = EI[(size_t)e * 2 + 1];

    float ef[EDGE_F];
#pragma unroll
    for (int k = 0; k < EDGE_F; ++k) ef[k] = EF[(size_t)e * EDGE_F + k];

    const int col = lane * 4;
    v4f t = *(const v4f*)(T + (size_t)src * OUT_F + col);   // b128 gather (L2 hit)

    float msg[4];
#pragma unroll
    for (int j = 0; j < 4; ++j) {
        float m = sbe[col + j];
#pragma unroll
        for (int k = 0; k < EDGE_F; ++k) m = fmaf(ef[k], sWe[k * OUT_F + col + j], m);
        msg[j] = m + t[j];
    }

    float* o = out + (size_t)dst * OUT_F + col;
#pragma unroll
    for (int j = 0; j < 4; ++j) atomicAdd(o + j, msg[j]);   // non-returning f32 atomic
}

// ---------------------------------------------------------------------------
// In-place ReLU on the accumulated output.
// ---------------------------------------------------------------------------
__global__ void gcn_relu(float* __restrict__ out, int n4) {
    int i = blockIdx.x * blockDim.x + threadIdx.x;
    if (i < n4) {
        v4f v = ((v4f*)out)[i];
#pragma unroll
        for (int j = 0; j < 4; ++j) v[j] = fmaxf(v[j], 0.f);
        ((v4f*)out)[i] = v;
    }
}

extern "C" void kernel_launch(void* const* d_in, const int* in_sizes, int n_in,
                              void* d_out, int out_size, void* d_ws, size_t ws_size,
                              hipStream_t stream) {
    const float* NF = (const float*)d_in[0];   // nodeFeatures [50000,128] f32
    const int*   EI = (const int*)  d_in[1];   // edgeIndex    [800000,2]  i32
    const float* EF = (const float*)d_in[2];   // edgeFeatures [800000,6]  f32
    const float* W  = (const float*)d_in[3];   // [128,128]
    const float* b  = (const float*)d_in[4];   // [128]
    const float* We = (const float*)d_in[5];   // [6,128]
    const float* be = (const float*)d_in[6];   // [128]
    float* out = (float*)d_out;                // [50000,128] f32
    float* T   = (float*)d_ws;                 // transformed nodes, 25.6 MB scratch

    const int n4 = (N_NODES * OUT_F) / 4;      // 1.6M float4
    gcn_zero<<<(n4 + 255) / 256, 256, 0, stream>>>(out, n4);
    gcn_node_transform<<<N_NODES / 16, 128, 0, stream>>>(NF, W, b, T);
    gcn_edge_scatter<<<(N_EDGES + 7) / 8, 256, 0, stream>>>(EI, EF, We, be, T, out);
    gcn_relu<<<(n4 + 255) / 256, 256, 0, stream>>>(out, n4);
}